// GAT_GCN_21457656611329
// MI455X (gfx1250) — compile-verified
//
#include <hip/hip_runtime.h>
#include <hip/hip_bf16.h>

// ---------------------------------------------------------------------------
// GAT + GCN + global pool + MLP forward for gfx1250 (MI455X, wave32, WMMA).
// GEMMs: V_WMMA_F32_16X16X4_F32 with LDS-staged tiles (branchless clamped
// loads -> no saveexec divergence; padded LDS strides for the 64-bank LDS).
// Edge scatters: block-per-edge, lane-per-feature (coalesced 185-float rows,
// f32 atomics into near-L2-resident accumulators).
// ---------------------------------------------------------------------------

typedef __attribute__((ext_vector_type(2))) float v2f;
typedef __attribute__((ext_vector_type(8))) float v8f;

#define NEG_SLOPE 0.2f

// Monotone float <-> uint encoding for atomicMax on floats (handles negatives).
__device__ __forceinline__ unsigned f2ord(float f) {
  unsigned u = __float_as_uint(f);
  return (u & 0x80000000u) ? ~u : (u | 0x80000000u);
}
__device__ __forceinline__ float ord2f(unsigned e) {
  unsigned u = (e & 0x80000000u) ? (e & 0x7FFFFFFFu) : ~e;
  return __uint_as_float(u);
}
#define ORD_NEG_INF 0x007FFFFFu  // f2ord(-INFINITY)

// ------------------------------- fills -------------------------------------
__global__ __launch_bounds__(256) void fill_f32(float* p, float v, size_t n) {
  size_t i = (size_t)blockIdx.x * blockDim.x + threadIdx.x;
  if (i < n) p[i] = v;
}
__global__ __launch_bounds__(256) void fill_u32(unsigned* p, unsigned v, size_t n) {
  size_t i = (size_t)blockIdx.x * blockDim.x + threadIdx.x;
  if (i < n) p[i] = v;
}

// --------------------------- WMMA f32 GEMM ---------------------------------
// C[M,Nc] = act(A[M,K] @ B[K,Nc] + bias).
// Block = 256 threads = 8 waves; tile = 128x16 (wave w owns rows 16w..16w+15).
// K chunked by 32 through LDS; B tile shared by all 8 waves (8x reuse).
#define MBLK 128
#define NBLK 16
#define KC   32

__global__ __launch_bounds__(256)
void wmma_gemm_f32(const float* __restrict__ A, const float* __restrict__ B,
                   const float* __restrict__ bias, float* __restrict__ C,
                   int M, int Nc, int K, int relu) {
  __shared__ float As[MBLK][KC + 1];   // stride 33: conflict-mitigated
  __shared__ float Bs[KC][NBLK + 1];   // stride 17: conflict-free
  const int tid   = threadIdx.x;       // 0..255
  const int wave  = tid >> 5;          // 0..7
  const int lane  = tid & 31;
  const int half  = lane >> 4;         // K sub-pair selector
  const int l     = lane & 15;
  const int tileM = blockIdx.x * MBLK;
  const int tileN = blockIdx.y * NBLK;

  v8f acc = {};
  for (int k0 = 0; k0 < K; k0 += KC) {
    // Stage A chunk (128x32): branchless clamped loads, coalesced along K.
    for (int i = tid; i < MBLK * KC; i += 256) {
      const int r = i >> 5, c = i & (KC - 1);
      const int gr = tileM + r, gk = k0 + c;
      const int cr = gr < M ? gr : M - 1;
      const int ck = gk < K ? gk : K - 1;
      const float v = A[(size_t)cr * K + ck];
      As[r][c] = (gr < M && gk < K) ? v : 0.0f;
    }
    // Stage B chunk (32x16).
    for (int i = tid; i < KC * NBLK; i += 256) {
      const int r = i >> 4, c = i & (NBLK - 1);
      const int gk = k0 + r, gc = tileN + c;
      const int ck = gk < K ? gk : K - 1;
      const int cc = gc < Nc ? gc : Nc - 1;
      const float v = B[(size_t)ck * Nc + cc];
      Bs[r][c] = (gk < K && gc < Nc) ? v : 0.0f;
    }
    __syncthreads();
    const int arow = wave * 16 + l;
#pragma unroll
    for (int kk = 0; kk < KC; kk += 4) {
      // A frag 16x4: lanes 0-15 hold K=kk,kk+1; lanes 16-31 K=kk+2,kk+3.
      v2f a, b;
      const int ka = kk + half * 2;
      a.x = As[arow][ka];
      a.y = As[arow][ka + 1];
      b.x = Bs[ka][l];
      b.y = Bs[ka + 1][l];
      acc = __builtin_amdgcn_wmma_f32_16x16x4_f32(
          /*neg_a=*/false, a, /*neg_b=*/false, b,
          /*c_mod=*/(short)0, acc, /*reuse_a=*/false, /*reuse_b=*/false);
    }
    __syncthreads();
  }
  // C/D layout: VGPR r -> row (16*wave + r) lanes 0-15, row (+8) lanes 16-31.
  const int col = tileN + l;
  if (col < Nc) {
    const float bv = bias ? bias[col] : 0.0f;
#pragma unroll
    for (int r = 0; r < 8; ++r) {
      const int row = tileM + wave * 16 + r + half * 8;
      if (row < M) {
        float v = acc[r] + bv;
        if (relu) v = fmaxf(v, 0.0f);
        C[(size_t)row * Nc + col] = v;
      }
    }
  }
}

// ---------------------- attention logit projections ------------------------
__global__ __launch_bounds__(256)
void attn_logits_kernel(const float* __restrict__ h, const float* __restrict__ a_src,
                        const float* __restrict__ a_dst, float* __restrict__ alS,
                        float* __restrict__ alD, int N, int H, int F) {
  size_t idx = (size_t)blockIdx.x * blockDim.x + threadIdx.x;
  if (idx >= (size_t)N * H) return;
  const int n = (int)(idx / H), hd = (int)(idx % H);
  const float* row = h + (size_t)n * H * F + (size_t)hd * F;
  const float* as = a_src + (size_t)hd * F;
  const float* ad = a_dst + (size_t)hd * F;
  float s = 0.0f, d = 0.0f;
  for (int f = 0; f < F; ++f) {
    const float v = row[f];
    s += v * as[f];
    d += v * ad[f];
  }
  alS[idx] = s;
  alD[idx] = d;
}

// --------------------------- edge passes -----------------------------------
__device__ __forceinline__ void edge_sd(const int* ei, int e, int E, int& s, int& d) {
  if (e < E) { s = ei[e]; d = ei[E + e]; }
  else       { s = e - E; d = e - E; }      // self loops appended
}

// Pass A: one thread per edge; all H heads. e=leakyrelu(...); segmax; deg.
__global__ __launch_bounds__(256)
void edge_logit_max_kernel(const int* __restrict__ ei, const float* __restrict__ alS,
                           const float* __restrict__ alD, float* __restrict__ ebuf,
                           unsigned* __restrict__ mU, float* __restrict__ deg,
                           int E, int Etot, int H) {
  const int e = blockIdx.x * blockDim.x + threadIdx.x;
  if (e >= Etot) return;
  int s, d; edge_sd(ei, e, E, s, d);
  for (int hd = 0; hd < H; ++hd) {
    float v = alS[(size_t)s * H + hd] + alD[(size_t)d * H + hd];
    v = (v > 0.0f) ? v : NEG_SLOPE * v;
    ebuf[(size_t)e * H + hd] = v;
    atomicMax(&mU[(size_t)d * H + hd], f2ord(v));
  }
  atomicAdd(&deg[d], 1.0f);
}

// m = isfinite(m) ? m : 0, decoded in place (uint -> float bits).
__global__ __launch_bounds__(256)
void finalize_m_kernel(unsigned* __restrict__ mU, size_t n) {
  size_t i = (size_t)blockIdx.x * blockDim.x + threadIdx.x;
  if (i >= n) return;
  float v = ord2f(mU[i]);
  if (!(v > -__builtin_inff() && v < __builtin_inff())) v = 0.0f;
  mU[i] = __float_as_uint(v);
}

// Pass B: ex = exp(e - m[dst]); stage back; segsum into denom.
__global__ __launch_bounds__(256)
void edge_exp_kernel(const int* __restrict__ ei, float* __restrict__ ebuf,
                     const float* __restrict__ mF, float* __restrict__ denom,
                     int E, int Etot, int H) {
  const int e = blockIdx.x * blockDim.x + threadIdx.x;
  if (e >= Etot) return;
  int s, d; edge_sd(ei, e, E, s, d); (void)s;
  for (int hd = 0; hd < H; ++hd) {
    const float ex = __expf(ebuf[(size_t)e * H + hd] - mF[(size_t)d * H + hd]);
    ebuf[(size_t)e * H + hd] = ex;
    atomicAdd(&denom[(size_t)d * H + hd], ex);
  }
}

// Pass C: alpha = ex / (denom[dst] + 1e-16).
__global__ __launch_bounds__(256)
void edge_alpha_kernel(const int* __restrict__ ei, float* __restrict__ ebuf,
                       const float* __restrict__ denom, int E, int Etot, int H) {
  const int e = blockIdx.x * blockDim.x + threadIdx.x;
  if (e >= Etot) return;
  int s, d; edge_sd(ei, e, E, s, d); (void)s;
  for (int hd = 0; hd < H; ++hd)
    ebuf[(size_t)e * H + hd] /= (denom[(size_t)d * H + hd] + 1e-16f);
}

// GAT aggregation: block per edge, lane per feature (192 covers HF=185).
__global__ __launch_bounds__(192)
void gat_agg_kernel(const int* __restrict__ ei, const float* __restrict__ alpha,
                    const float* __restrict__ h, float* __restrict__ agg,
                    int E, int HF, int F, int H) {
  __shared__ float al[8];
  const int e = blockIdx.x;
  int s, d; edge_sd(ei, e, E, s, d);
  if (threadIdx.x < H) al[threadIdx.x] = alpha[(size_t)e * H + threadIdx.x];
  __syncthreads();
  const int f = threadIdx.x;
  if (f < HF) {
    int hd = 0, base = F;                 // head index: <=4 cheap compares
    while (f >= base) { base += F; ++hd; }
    atomicAdd(&agg[(size_t)d * HF + f], al[hd] * h[(size_t)s * HF + f]);
  }
}

// x = relu(x + bias[f]) in place; block per node.
__global__ __launch_bounds__(192)
void bias_relu_kernel(float* __restrict__ p, const float* __restrict__ bias, int HF) {
  const int f = threadIdx.x;
  if (f < HF) {
    const size_t i = (size_t)blockIdx.x * HF + f;
    p[i] = fmaxf(p[i] + bias[f], 0.0f);
  }
}

__global__ __launch_bounds__(256)
void dinv_kernel(const float* __restrict__ deg, float* __restrict__ dinv, int N) {
  size_t i = (size_t)blockIdx.x * blockDim.x + threadIdx.x;
  if (i < (size_t)N) dinv[i] = rsqrtf(fmaxf(deg[i], 1.0f));
}

// GCN aggregation: block per edge, lane per feature.
__global__ __launch_bounds__(192)
void gcn_agg_kernel(const int* __restrict__ ei, const float* __restrict__ h2,
                    const float* __restrict__ dinv, float* __restrict__ x2,
                    int E, int HF) {
  const int e = blockIdx.x;
  int s, d; edge_sd(ei, e, E, s, d);
  const float nrm = dinv[s] * dinv[d];
  const int f = threadIdx.x;
  if (f < HF)
    atomicAdd(&x2[(size_t)d * HF + f], nrm * h2[(size_t)s * HF + f]);
}

// Global pooling: block per node, lane per feature.
__global__ __launch_bounds__(192)
void pool_kernel(const float* __restrict__ x2, const int* __restrict__ batch,
                 float* __restrict__ gsum, unsigned* __restrict__ gmaxU,
                 float* __restrict__ cnt, int HF) {
  const int n = blockIdx.x;
  const int g = batch[n];
  const int f = threadIdx.x;
  if (f < HF) {
    const float v = x2[(size_t)n * HF + f];
    atomicAdd(&gsum[(size_t)g * HF + f], v);
    atomicMax(&gmaxU[(size_t)g * HF + f], f2ord(v));
  }
  if (threadIdx.x == 0) atomicAdd(&cnt[g], 1.0f);
}

// g = [gmax | gmean]; block per graph, lane per column (384 covers 370).
__global__ __launch_bounds__(384)
void build_gcat_kernel(const float* __restrict__ gsum, const unsigned* __restrict__ gmaxU,
                       const float* __restrict__ cnt, float* __restrict__ gcat, int HF) {
  const int g = blockIdx.x;
  const int c = threadIdx.x;
  const int W = 2 * HF;
  if (c >= W) return;
  const float cg = cnt[g];
  float v;
  if (c < HF) v = (cg > 0.0f) ? ord2f(gmaxU[(size_t)g * HF + c]) : 0.0f;
  else        v = gsum[(size_t)g * HF + (c - HF)] / fmaxf(cg, 1.0f);
  gcat[(size_t)g * W + c] = v;
}

// ---------------------------------------------------------------------------
extern "C" void kernel_launch(void* const* d_in, const int* in_sizes, int n_in,
                              void* d_out, int out_size, void* d_ws, size_t ws_size,
                              hipStream_t stream) {
  (void)n_in; (void)out_size; (void)ws_size;
  const float* x     = (const float*)d_in[0];
  const int*   ei    = (const int*)  d_in[1];
  const int*   batch = (const int*)  d_in[2];
  const float* W_gat = (const float*)d_in[3];
  const float* a_src = (const float*)d_in[4];
  const float* a_dst = (const float*)d_in[5];
  const float* b_gat = (const float*)d_in[6];
  const float* W_gcn = (const float*)d_in[7];
  const float* b_gcn = (const float*)d_in[8];
  const float* Wg1   = (const float*)d_in[9];
  const float* bg1   = (const float*)d_in[10];
  const float* W1    = (const float*)d_in[11];
  const float* b1    = (const float*)d_in[12];
  const float* Wo    = (const float*)d_in[13];
  const float* bo    = (const float*)d_in[14];

  const int N    = in_sizes[2];          // 262144
  const int E    = in_sizes[1] / 2;      // 1048576
  const int H    = 5;
  const int F    = in_sizes[4] / H;      // 37
  const int HF   = H * F;                // 185
  const int Etot = E + N;                // self loops appended
  const int G    = 8192;

  float* out_head = (float*)d_out;       // [G,1]
  float* alpha    = out_head + G;        // [Etot,H], staged in place

  // ---- carve workspace (256B aligned) ----
  char* base = (char*)d_ws;
  size_t off = 0;
  auto carve = [&](size_t bytes) -> char* {
    char* p = base + off;
    off += (bytes + 255) & ~(size_t)255;
    return p;
  };
  float*    big0  = (float*)   carve((size_t)N * HF * 4);  // h, then h2
  float*    big1  = (float*)   carve((size_t)N * HF * 4);  // agg/x1, then x2
  float*    alS   = (float*)   carve((size_t)N * H * 4);
  float*    alD   = (float*)   carve((size_t)N * H * 4);
  unsigned* mU    = (unsigned*)carve((size_t)N * H * 4);   // -> float m in place
  float*    denom = (float*)   carve((size_t)N * H * 4);
  float*    deg   = (float*)   carve((size_t)N * 4);
  float*    dinv  = (float*)   carve((size_t)N * 4);
  float*    gsum  = (float*)   carve((size_t)G * HF * 4);
  unsigned* gmaxU = (unsigned*)carve((size_t)G * HF * 4);
  float*    cnt   = (float*)   carve((size_t)G * 4);
  float*    gcat  = (float*)   carve((size_t)G * 2 * HF * 4);
  float*    mlp1  = (float*)   carve((size_t)G * 256 * 4);
  float*    mlp2  = (float*)   carve((size_t)G * 512 * 4);

  const int TB = 256;
  auto nb = [&](size_t n) -> unsigned { return (unsigned)((n + TB - 1) / TB); };
  auto gemm_grid = [&](int M, int Nc) {
    return dim3((M + MBLK - 1) / MBLK, (Nc + NBLK - 1) / NBLK);
  };

  // 1) h = x @ W_gat                                    [N,37]@[37,185]
  wmma_gemm_f32<<<gemm_grid(N, HF), 256, 0, stream>>>(x, W_gat, nullptr, big0,
                                                      N, HF, F, 0);
  // 2) attention logit projections
  attn_logits_kernel<<<nb((size_t)N * H), TB, 0, stream>>>(big0, a_src, a_dst,
                                                           alS, alD, N, H, F);
  // 3) init segment buffers
  fill_u32<<<nb((size_t)N * H), TB, 0, stream>>>(mU, ORD_NEG_INF, (size_t)N * H);
  fill_f32<<<nb((size_t)N * H), TB, 0, stream>>>(denom, 0.0f, (size_t)N * H);
  fill_f32<<<nb((size_t)N), TB, 0, stream>>>(deg, 0.0f, (size_t)N);

  // 4) segment softmax over incoming edges (3 passes, alpha staged in d_out)
  edge_logit_max_kernel<<<nb((size_t)Etot), TB, 0, stream>>>(ei, alS, alD, alpha,
                                                             mU, deg, E, Etot, H);
  finalize_m_kernel<<<nb((size_t)N * H), TB, 0, stream>>>(mU, (size_t)N * H);
  edge_exp_kernel<<<nb((size_t)Etot), TB, 0, stream>>>(ei, alpha, (const float*)mU,
                                                       denom, E, Etot, H);
  edge_alpha_kernel<<<nb((size_t)Etot), TB, 0, stream>>>(ei, alpha, denom, E, Etot, H);

  // 5) GAT aggregation + bias/relu -> x1 (in big1)
  fill_f32<<<nb((size_t)N * HF), TB, 0, stream>>>(big1, 0.0f, (size_t)N * HF);
  gat_agg_kernel<<<Etot, 192, 0, stream>>>(ei, alpha, big0, big1, E, HF, F, H);
  bias_relu_kernel<<<N, 192, 0, stream>>>(big1, b_gat, HF);

  // 6) symmetric normalization factors
  dinv_kernel<<<nb((size_t)N), TB, 0, stream>>>(deg, dinv, N);

  // 7) h2 = x1 @ W_gcn                                  [N,185]@[185,185]
  wmma_gemm_f32<<<gemm_grid(N, HF), 256, 0, stream>>>(big1, W_gcn, nullptr, big0,
                                                      N, HF, HF, 0);

  // 8) GCN aggregation + bias/relu -> x2 (in big1)
  fill_f32<<<nb((size_t)N * HF), TB, 0, stream>>>(big1, 0.0f, (size_t)N * HF);
  gcn_agg_kernel<<<Etot, 192, 0, stream>>>(ei, big0, dinv, big1, E, HF);
  bias_relu_kernel<<<N, 192, 0, stream>>>(big1, b_gcn, HF);

  // 9) global max+mean pooling
  fill_f32<<<nb((size_t)G * HF), TB, 0, stream>>>(gsum, 0.0f, (size_t)G * HF);
  fill_u32<<<nb((size_t)G * HF), TB, 0, stream>>>(gmaxU, ORD_NEG_INF, (size_t)G * HF);
  fill_f32<<<nb((size_t)G), TB, 0, stream>>>(cnt, 0.0f, (size_t)G);
  pool_kernel<<<N, 192, 0, stream>>>(big1, batch, gsum, gmaxU, cnt, HF);
  build_gcat_kernel<<<G, 384, 0, stream>>>(gsum, gmaxU, cnt, gcat, HF);

  // 10) MLP head (WMMA), final scores into d_out[0:G]
  wmma_gemm_f32<<<gemm_grid(G, 256), 256, 0, stream>>>(gcat, Wg1, bg1, mlp1,
                                                       G, 256, 2 * HF, 1);
  wmma_gemm_f32<<<gemm_grid(G, 512), 256, 0, stream>>>(mlp1, W1, b1, mlp2,
                                                       G, 512, 256, 1);
  wmma_gemm_f32<<<gemm_grid(G, 1), 256, 0, stream>>>(mlp2, Wo, bo, out_head,
                                                     G, 1, 512, 0);
}